// Gabor2d_67542655697161
// MI455X (gfx1250) — compile-verified
//
#include <hip/hip_runtime.h>

typedef __attribute__((ext_vector_type(2))) float v2f;
typedef __attribute__((ext_vector_type(8))) float v8f;

#define IN_H     224
#define IN_W     224
#define OUT_HW   28
#define TAPS     169   // 13*13 common padded tap grid
#define TAPS_PAD 172   // round up to multiple of 4 (K per WMMA)
#define WSTRIDE  176   // per-filter LDS stride (zero tail covers taps 169..175)

// One block = one (image, output row). 64 threads = 2 waves.
// Wave w computes output pixels ox = 16w .. 16w+15 (only ox<28 stored).
__global__ __launch_bounds__(64) void Gabor2d_wmma_kernel(
    const float* __restrict__ x,
    const float* __restrict__ k0, const float* __restrict__ k1,
    const float* __restrict__ k2, const float* __restrict__ k3,
    float* __restrict__ out)
{
    __shared__ float Wlds[4 * WSTRIDE];   // 4 filters padded onto 13x13 grid
    __shared__ float tile[2][16][16];     // per-wave 16(M pixels) x 16(N filters) D tile

    const int tid  = threadIdx.x;         // 0..63
    const int wave = tid >> 5;            // 0..1
    const int lane = tid & 31;
    const int b    = blockIdx.x / OUT_HW; // image index
    const int oy   = blockIdx.x % OUT_HW; // output row

    // ---- stage zero-padded weights into LDS (11 entries per thread) ----
    for (int e = tid; e < 4 * WSTRIDE; e += 64) {
        const int n = e / WSTRIDE;        // filter 0..3
        const int t = e % WSTRIDE;        // tap on 13x13 grid (+ pad tail)
        float w = 0.0f;
        if (t < TAPS) {
            const int dy = t / 13 - 6;
            const int dx = t % 13 - 6;
            const int p  = 3 + n;         // radii 3,4,5,6
            const int ks = 7 + 2 * n;     // sizes 7,9,11,13
            if (dy >= -p && dy <= p && dx >= -p && dx <= p) {
                const float* kp = (n == 0) ? k0 : (n == 1) ? k1 : (n == 2) ? k2 : k3;
                w = kp[(dy + p) * ks + (dx + p)];
            }
        }
        Wlds[e] = w;
    }
    __syncthreads();

    // ---- implicit GEMM over taps: D(16 px x 16 filt) += A(16x4) * B(4x16) ----
    const int m   = lane & 15;            // A: pixel row M;  B: column N
    const int hi  = lane >> 4;            // 0 -> taps K+0/K+1, 1 -> taps K+2/K+3
    const int ox  = wave * 16 + m;        // output column (>=28 lanes are padding)
    const int cy  = oy * 8;               // input center row
    const int cx  = ox * 8;               // input center col
    const float* __restrict__ xb = x + (size_t)b * (IN_H * IN_W);

    v8f acc = {};
    for (int kk = 0; kk < TAPS_PAD; kk += 4) {
        const int t0 = kk + 2 * hi;
        const int t1 = t0 + 1;

        // A fragment: gathered patch values for pixel m at taps t0,t1 (OOB -> 0)
        float a0 = 0.0f, a1 = 0.0f;
        if (t0 < TAPS) {
            const int iy = cy + t0 / 13 - 6;
            const int ix = cx + t0 % 13 - 6;
            if (iy >= 0 && iy < IN_H && ix >= 0 && ix < IN_W) a0 = xb[iy * IN_W + ix];
        }
        if (t1 < TAPS) {
            const int iy = cy + t1 / 13 - 6;
            const int ix = cx + t1 % 13 - 6;
            if (iy >= 0 && iy < IN_H && ix >= 0 && ix < IN_W) a1 = xb[iy * IN_W + ix];
        }
        v2f a; a.x = a0; a.y = a1;

        // B fragment: filter weights; columns 4..15 are zero
        float b0 = 0.0f, b1 = 0.0f;
        if (m < 4) {
            b0 = Wlds[m * WSTRIDE + t0];
            b1 = Wlds[m * WSTRIDE + t1];
        }
        v2f bf; bf.x = b0; bf.y = b1;

        // (neg_a, A, neg_b, B, c_mod, C, reuse_a, reuse_b)
        acc = __builtin_amdgcn_wmma_f32_16x16x4_f32(false, a, false, bf,
                                                    (short)0, acc, false, false);
    }

    // ---- spill D tile: VGPR r, lanes<16 -> M=r, lanes>=16 -> M=8+r; N = lane&15 ----
    #pragma unroll
    for (int r = 0; r < 8; ++r) {
        tile[wave][hi * 8 + r][m] = acc[r];
    }
    __syncthreads();

    // ---- pairwise max of adjacent filters, replicate over 4 theta groups ----
    // out layout: [b][theta(4)][pair(3)][28][28]
    if (ox < OUT_HW) {
        const float c0 = tile[wave][m][0];
        const float c1 = tile[wave][m][1];
        const float c2 = tile[wave][m][2];
        const float c3 = tile[wave][m][3];
        const float pr0 = fmaxf(c0, c1);
        const float pr1 = fmaxf(c1, c2);
        const float pr2 = fmaxf(c2, c3);
        const size_t plane = (size_t)OUT_HW * OUT_HW;                 // 784
        const size_t base  = (size_t)b * (4 * 3 * plane)
                           + (size_t)oy * OUT_HW + (size_t)ox;
        // split the 4 theta copies across the two lane halves (2 stores each)
        #pragma unroll
        for (int th = 0; th < 2; ++th) {
            const size_t tb = base + ((size_t)(2 * th + hi) * 3) * plane;
            out[tb + 0 * plane] = pr0;
            out[tb + 1 * plane] = pr1;
            out[tb + 2 * plane] = pr2;
        }
    }
}

extern "C" void kernel_launch(void* const* d_in, const int* in_sizes, int n_in,
                              void* d_out, int out_size, void* d_ws, size_t ws_size,
                              hipStream_t stream) {
    (void)in_sizes; (void)n_in; (void)d_ws; (void)ws_size; (void)out_size;
    const float* x  = (const float*)d_in[0];  // [64,1,224,224]
    const float* k0 = (const float*)d_in[1];  // [1,1,7,7]
    const float* k1 = (const float*)d_in[2];  // [1,1,9,9]
    const float* k2 = (const float*)d_in[3];  // [1,1,11,11]
    const float* k3 = (const float*)d_in[4];  // [1,1,13,13]
    float* out = (float*)d_out;               // [64, 4*3*28*28]

    const int blocks = 64 * OUT_HW;           // one block per (image, output row)
    Gabor2d_wmma_kernel<<<blocks, 64, 0, stream>>>(x, k0, k1, k2, k3, out);
}